// Frequency_filter_67628555043025
// MI455X (gfx1250) — compile-verified
//
#include <hip/hip_runtime.h>
#include <hip/hip_bf16.h>

typedef __attribute__((ext_vector_type(2))) float v2f;
typedef __attribute__((ext_vector_type(8))) float v8f;

#define NN 8
#define CC 64
#define HH 128
#define WW 128
#define GK2 72
#define FS 40   // fused LDS stride: 2*40 % 64 == 16 -> lane halves hit disjoint banks
#define OWS 80  // out_w^T LDS stride: 2*80 % 64 == 32 -> disjoint banks

// ---------------- Kernel 1: global average pool (gap) ----------------
__global__ void ff_gap_kernel(const float* __restrict__ x, float* __restrict__ gap) {
    __shared__ float sd[256];
    int nc = blockIdx.x;                       // 0..511 = n*64+c
    const float* p = x + (size_t)nc * (HH * WW);
    float s = 0.f;
    for (int i = threadIdx.x; i < HH * WW; i += 256) s += p[i];
    sd[threadIdx.x] = s;
    __syncthreads();
    for (int off = 128; off > 0; off >>= 1) {
        if (threadIdx.x < off) sd[threadIdx.x] += sd[threadIdx.x + off];
        __syncthreads();
    }
    if (threadIdx.x == 0) gap[nc] = sd[0] * (1.0f / (HH * WW));
}

// ---------------- Kernel 2: tiny MLPs + softmaxes (one block) ----------------
__global__ void ff_small_kernel(const float* __restrict__ gap_g,
                                const float* __restrict__ conv_w,
                                const float* __restrict__ bn_gamma,
                                const float* __restrict__ bn_beta,
                                const float* __restrict__ bn_mean,
                                const float* __restrict__ bn_var,
                                const float* __restrict__ fc_w,
                                const float* __restrict__ fc_b,
                                const float* __restrict__ s0_w,
                                const float* __restrict__ s0_b,
                                const float* __restrict__ s1_w,
                                const float* __restrict__ s1_b,
                                float* __restrict__ lf_out,
                                float* __restrict__ ha_out,
                                float* __restrict__ la_out) {
    __shared__ float gapS[NN][CC];
    __shared__ float lfS[NN][GK2];
    __shared__ float fcvS[NN][32];
    __shared__ float attS[NN][128];
    int tid = threadIdx.x;

    for (int i = tid; i < NN * CC; i += 256) gapS[i >> 6][i & 63] = gap_g[i];
    __syncthreads();

    // lf = gap @ conv_w^T, then BN
    for (int t = tid; t < NN * GK2; t += 256) {
        int n = t / GK2, j = t - n * GK2;
        float d = 0.f;
        for (int c = 0; c < CC; ++c) d += gapS[n][c] * conv_w[j * CC + c];
        d = (d - bn_mean[j]) * rsqrtf(bn_var[j] + 1e-5f) * bn_gamma[j] + bn_beta[j];
        lfS[n][j] = d;
    }
    // fcv = gap @ fc_w^T + fc_b   (8*32 == 256 threads, one each)
    {
        int n = tid >> 5, j = tid & 31;
        float d = 0.f;
        for (int c = 0; c < CC; ++c) d += gapS[n][c] * fc_w[j * CC + c];
        fcvS[n][j] = d + fc_b[j];
    }
    __syncthreads();

    // softmax over k2=9 per (n, group)
    if (tid < NN * 8) {
        int n = tid >> 3, g = tid & 7;
        float m = -1e30f;
        for (int k = 0; k < 9; ++k) m = fmaxf(m, lfS[n][g * 9 + k]);
        float e[9], ssum = 0.f;
        for (int k = 0; k < 9; ++k) { e[k] = expf(lfS[n][g * 9 + k] - m); ssum += e[k]; }
        float inv = 1.f / ssum;
        for (int k = 0; k < 9; ++k) lf_out[(n * 8 + g) * 9 + k] = e[k] * inv;
    }
    // ha/la raw
    for (int t = tid; t < NN * 128; t += 256) {
        int n = t >> 7, i = t & 127;
        const float* wrow;
        float bias;
        if (i < 64) { wrow = s0_w + i * 32;        bias = s0_b[i]; }
        else        { wrow = s1_w + (i - 64) * 32; bias = s1_b[i - 64]; }
        float d = 0.f;
        for (int j = 0; j < 32; ++j) d += fcvS[n][j] * wrow[j];
        attS[n][i] = d + bias;
    }
    __syncthreads();

    // softmax over concat(ha,la) of length 128 per n
    if (tid < NN) {
        int n = tid;
        float m = -1e30f;
        for (int i = 0; i < 128; ++i) m = fmaxf(m, attS[n][i]);
        float ssum = 0.f;
        for (int i = 0; i < 128; ++i) ssum += expf(attS[n][i] - m);
        float inv = 1.f / ssum;
        for (int i = 0; i < 64; ++i) ha_out[n * 64 + i] = expf(attS[n][i] - m) * inv;
        for (int i = 0; i < 64; ++i) la_out[n * 64 + i] = expf(attS[n][64 + i] - m) * inv;
    }
}

// ---------------- Kernel 3: filter + fuse + 64x64 channel GEMM (WMMA fp32) ----------------
// grid: (4 col-tiles, 128 rows, 8 samples), block 256 (8 waves)
__global__ void __launch_bounds__(256)
ff_main_kernel(const float* __restrict__ x,
               const float* __restrict__ lf,   // [n][8][9] post-softmax
               const float* __restrict__ ha,   // [n][64]
               const float* __restrict__ la,   // [n][64]
               const float* __restrict__ out_w,// [o][c] 64x64
               const float* __restrict__ out_b,// [64]
               float* __restrict__ out) {
    __shared__ float xs[3][CC][34];       // 3 rows x 64 ch x (32 + 2 halo)
    __shared__ float fusedS[CC][FS];      // B matrix: [cin][pixel]
    __shared__ float owT[CC][OWS];        // A matrix transposed: owT[cin][cout]
    __shared__ float lfS[GK2];
    __shared__ float haS[CC], laS[CC], obS[CC];

    const int tid = threadIdx.x;
    const int n = blockIdx.z, h = blockIdx.y;
    const int w0 = blockIdx.x * 32;

    int rows[3];
    rows[0] = (h == 0) ? 1 : h - 1;            // reflect pad
    rows[1] = h;
    rows[2] = (h == HH - 1) ? HH - 2 : h + 1;

    // ---- stage 1: stage inputs into LDS ----
    for (int idx = tid; idx < 3 * CC * 34; idx += 256) {
        int r = idx / (CC * 34);
        int rem = idx - r * (CC * 34);
        int c = rem / 34;
        int j = rem - c * 34;
        int wc = w0 - 1 + j;
        wc = (wc < 0) ? 1 : ((wc > WW - 1) ? WW - 2 : wc);  // reflect at image edge
        xs[r][c][j] = x[(((n * CC + c) * HH) + rows[r]) * WW + wc];
    }
    for (int idx = tid; idx < CC * CC; idx += 256) {
        int o = idx >> 6, c = idx & 63;
        owT[c][o] = out_w[idx];               // transpose: A[k][m]
    }
    if (tid < GK2) lfS[tid] = lf[n * GK2 + tid];
    if (tid < CC) {
        haS[tid] = ha[n * CC + tid];
        laS[tid] = la[n * CC + tid];
        obS[tid] = out_b[tid];
    }
    __syncthreads();

    // ---- stage 2: fused[c][w] = ha*x + (la-ha)*low ----
    for (int idx = tid; idx < CC * 32; idx += 256) {
        int c = idx >> 5, j = idx & 31;       // j = local pixel; xs col j+1 is center
        const float* l9 = &lfS[(c >> 3) * 9];
        float low = 0.f;
#pragma unroll
        for (int dh = 0; dh < 3; ++dh)
#pragma unroll
            for (int dw = 0; dw < 3; ++dw)
                low += l9[dh * 3 + dw] * xs[dh][c][j + dw];
        float xc = xs[1][c][j + 1];
        fusedS[c][j] = haS[c] * xc + (laS[c] - haS[c]) * low;
    }
    __syncthreads();

    // ---- stage 3: out[64 x 32] = out_w[64x64] @ fused[64x32] via v_wmma_f32_16x16x4_f32 ----
    const int wave = tid >> 5;
    const int lane = tid & 31;
    const int half = lane >> 4;               // K-pair selector per ISA layout
    const int l15  = lane & 15;
    const int mt = wave & 3;                  // 4 M-tiles of 16 out-channels
    const int nt = wave >> 2;                 // 2 N-tiles of 16 pixels

    v8f acc = {};
#pragma unroll
    for (int kk = 0; kk < CC; kk += 4) {
        int k = kk + 2 * half;
        v2f a, b;
        int row = mt * 16 + l15;              // A: M = lane%16 in both halves
        a.x = owT[k][row];
        a.y = owT[k + 1][row];
        int col = nt * 16 + l15;              // B: N = lane%16 in both halves
        b.x = fusedS[k][col];
        b.y = fusedS[k + 1][col];
        acc = __builtin_amdgcn_wmma_f32_16x16x4_f32(
            false, a, false, b, (short)0, acc, false, false);
    }
#pragma unroll
    for (int r = 0; r < 8; ++r) {
        int o = mt * 16 + 8 * half + r;       // D: VGPR r -> M = r + 8*half
        int wcol = w0 + nt * 16 + l15;
        out[(((n * CC + o) * HH) + h) * WW + wcol] = acc[r] + obS[o];
    }
}

extern "C" void kernel_launch(void* const* d_in, const int* in_sizes, int n_in,
                              void* d_out, int out_size, void* d_ws, size_t ws_size,
                              hipStream_t stream) {
    (void)in_sizes; (void)n_in; (void)out_size; (void)ws_size;
    const float* x        = (const float*)d_in[0];
    const float* conv_w   = (const float*)d_in[1];
    const float* bn_gamma = (const float*)d_in[2];
    const float* bn_beta  = (const float*)d_in[3];
    const float* bn_mean  = (const float*)d_in[4];
    const float* bn_var   = (const float*)d_in[5];
    const float* fc_w     = (const float*)d_in[6];
    const float* fc_b     = (const float*)d_in[7];
    const float* s0_w     = (const float*)d_in[8];
    const float* s0_b     = (const float*)d_in[9];
    const float* s1_w     = (const float*)d_in[10];
    const float* s1_b     = (const float*)d_in[11];
    const float* out_w    = (const float*)d_in[12];
    const float* out_b    = (const float*)d_in[13];

    float* ws  = (float*)d_ws;
    float* gap = ws;            // 512
    float* lfp = ws + 512;      // 576
    float* hap = ws + 1088;     // 512
    float* lap = ws + 1600;     // 512

    ff_gap_kernel<<<NN * CC, 256, 0, stream>>>(x, gap);
    ff_small_kernel<<<1, 256, 0, stream>>>(gap, conv_w, bn_gamma, bn_beta, bn_mean,
                                           bn_var, fc_w, fc_b, s0_w, s0_b, s1_w, s1_b,
                                           lfp, hap, lap);
    dim3 grid(WW / 32, HH, NN);
    ff_main_kernel<<<grid, 256, 0, stream>>>(x, lfp, hap, lap, out_w, out_b,
                                             (float*)d_out);
}